// MoEGate_24902220382973
// MI455X (gfx1250) — compile-verified
//
#include <hip/hip_runtime.h>

typedef float v2f __attribute__((ext_vector_type(2)));
typedef float v8f __attribute__((ext_vector_type(8)));
typedef unsigned int u32x4 __attribute__((ext_vector_type(4)));
typedef int i32x4 __attribute__((ext_vector_type(4)));
typedef int i32x8 __attribute__((ext_vector_type(8)));

#define HDIM 2048
#define NEXP 256
#define BM   64           // tokens per block (gemm)
#define BK   32           // K chunk (floats)
#define LDSTRIDE (BK + 4) // 36 floats: distinct banks across 16 rows, 16B-aligned rows
#define NITER (HDIM / BK) // 64

// ---------------------------------------------------------------------------
// TDM 2-D tile load: global[row-major, stride 2048 floats] -> LDS, with the
// +4-DWORD row padding done by the DMA engine (pad_interval=32 DW, pad=4 DW).
// Descriptor per CDNA5 ISA. Issued by one wave; tracked
// on TENSORcnt. This toolchain exposes the 6-arg builtin:
//   (u32x4 g0, i32x8 g1, i32x4 g2, i32x4 g3, i32x8 extra, i32 cpol)
// groups 2/3 + extra are zero (2-D tile, tile_dim2=0).
// ---------------------------------------------------------------------------
__device__ __forceinline__ void tdm_load_2d(const float* gptr, unsigned lds_byte,
                                            unsigned tensor_d1, unsigned tile_d1) {
  unsigned long long ga = (unsigned long long)(uintptr_t)gptr;
  u32x4 g0;
  g0[0] = 1u;                                    // count=1 valid descriptor
  g0[1] = lds_byte;                              // lds_addr
  g0[2] = (unsigned)ga;                          // global_addr[31:0]
  g0[3] = (unsigned)(ga >> 32) | 0x80000000u;    // global_addr[56:32] | type=2
  i32x8 g1;
  g1[0] = (2 << 16)        // data_size = 4B
        | (1 << 20)        // pad_enable
        | (4 << 22)        // pad_interval: 32 DWORDs
        | (3 << 25);       // pad_amount: 4 DWORDs
  g1[1] = (int)((HDIM & 0xffffu) << 16);         // tensor_dim0 low16 @ bits 63:48
  g1[2] = (int)(tensor_d1 << 16);                // dim0 hi16=0 | tensor_dim1 low16
  g1[3] = (int)((unsigned)BK << 16);             // dim1 hi16=0 | tile_dim0=32
  g1[4] = (int)tile_d1;                          // tile_dim1 | tile_dim2=0
  g1[5] = HDIM;                                  // tensor_dim0_stride low32
  g1[6] = 0;                                     // stride hi16 | dim1_stride low16
  g1[7] = 0;                                     // dim1_stride hi32
  i32x4 z4 = {0, 0, 0, 0};                       // groups 2/3 unused (2-D tile)
  i32x8 z8 = {0, 0, 0, 0, 0, 0, 0, 0};
  __builtin_amdgcn_tensor_load_to_lds(g0, g1, z4, z4, z8, 0);
}

// ---------------------------------------------------------------------------
// Kernel 1: logits = hidden @ weight^T via V_WMMA_F32_16X16X4_F32 (exact fp32)
// 512 threads = 16 waves; wave w owns expert tile w; double-buffered LDS fed
// by the Tensor Data Mover, synced with s_wait_tensorcnt + barriers.
// ---------------------------------------------------------------------------
__global__ __launch_bounds__(512) void moe_gemm_logits(
    const float* __restrict__ hidden, const float* __restrict__ weight,
    float* __restrict__ logits) {
  __shared__ float As[2 * BM * LDSTRIDE];    // 2 x 64 x 36 floats (18 KB)
  __shared__ float Bs[2 * NEXP * LDSTRIDE];  // 2 x 256 x 36 floats (72 KB)

  const int tid  = threadIdx.x;
  const int lane = tid & 31;
  const int wave = tid >> 5;                 // 0..15 -> expert tile
  const int tokBase = blockIdx.x * BM;

  v8f acc[4] = {};                           // 4 token tiles x (16x16 f32 D)

  const int mn   = lane & 15;                // M for A-frag, N for B-frag
  const int koff = (lane >> 4) * 2;          // K sub-offset per WMMA layout

  const bool leader = (wave == 0);           // wave-uniform scalar branch
  const unsigned asBase = (unsigned)(uintptr_t)(void*)As;
  const unsigned bsBase = (unsigned)(uintptr_t)(void*)Bs;

  if (leader) {  // prologue: DMA buffer 0 (chunk 0): A tile + B tile
    tdm_load_2d(hidden + (size_t)tokBase * HDIM, asBase, 16384u, BM);
    tdm_load_2d(weight, bsBase, NEXP, NEXP);
  }

  for (int i = 0; i < NITER; ++i) {
    const int cur = i & 1;
    __syncthreads();  // buffer (1-cur) free: compute of iter i-1 finished
    if (leader) {
      if (i + 1 < NITER) {
        const int k1 = (i + 1) * BK;
        tdm_load_2d(hidden + (size_t)tokBase * HDIM + k1,
                    asBase + (1 - cur) * BM * LDSTRIDE * 4, 16384u, BM);
        tdm_load_2d(weight + k1,
                    bsBase + (1 - cur) * NEXP * LDSTRIDE * 4, NEXP, NEXP);
        // TENSORcnt completes in order: <=2 outstanding => buffer `cur` done.
        __builtin_amdgcn_s_wait_tensorcnt(2);
      } else {
        __builtin_amdgcn_s_wait_tensorcnt(0);
      }
    }
    __syncthreads();  // buffer `cur` visible to all waves

    const float* as = As + cur * BM * LDSTRIDE;
    const float* bs = Bs + cur * NEXP * LDSTRIDE;
#pragma unroll
    for (int kk = 0; kk < BK; kk += 4) {
      v2f b = *reinterpret_cast<const v2f*>(
          bs + (wave * 16 + mn) * LDSTRIDE + kk + koff);
#pragma unroll
      for (int t = 0; t < 4; ++t) {
        v2f a = *reinterpret_cast<const v2f*>(
            as + (t * 16 + mn) * LDSTRIDE + kk + koff);
        acc[t] = __builtin_amdgcn_wmma_f32_16x16x4_f32(
            false, a, false, b, (short)0, acc[t], false, false);
      }
    }
  }

  // store D per documented layout: VGPR r -> M=r (lanes 0-15) / M=r+8 (16-31)
  const int n   = wave * 16 + (lane & 15);
  const int mhi = (lane >> 4) * 8;
#pragma unroll
  for (int t = 0; t < 4; ++t) {
#pragma unroll
    for (int r = 0; r < 8; ++r) {
      logits[(size_t)(tokBase + t * 16 + r + mhi) * NEXP + n] = acc[t][r];
    }
  }
}

// ---------------------------------------------------------------------------
// Kernel 2: per-token grouped top-k + softmax/count accumulation.
// One wave32 per token (group_size 32 == wave32). 8 waves/block, 4 tokens/wave.
// ---------------------------------------------------------------------------
__device__ __forceinline__ float waveMax32(float v) {
#pragma unroll
  for (int off = 16; off > 0; off >>= 1)
    v = fmaxf(v, __shfl_xor(v, off, 32));
  return v;
}
__device__ __forceinline__ float waveSum32(float v) {
#pragma unroll
  for (int off = 16; off > 0; off >>= 1)
    v += __shfl_xor(v, off, 32);
  return v;
}

__global__ __launch_bounds__(256) void moe_gate_topk(
    const float* __restrict__ logits, float* __restrict__ out_idx,
    float* __restrict__ out_w, float* __restrict__ g_cnt,
    float* __restrict__ g_prb, int n_tokens) {
  __shared__ float s_prb[NEXP];
  __shared__ float s_cnt[NEXP];
  const int tid = threadIdx.x;
  const int lane = tid & 31;
  const int wave = tid >> 5;

  s_prb[tid] = 0.0f;
  s_cnt[tid] = 0.0f;
  __syncthreads();

  for (int t = 0; t < 4; ++t) {
    int tok = blockIdx.x * 32 + wave * 4 + t;
    if (tok >= n_tokens) continue;
    const float* lg = logits + (size_t)tok * NEXP;

    float v[8];
#pragma unroll
    for (int g = 0; g < 8; ++g) v[g] = lg[g * 32 + lane];

    // --- softmax over all 256 logits, accumulate mean probs ---
    float mx = v[0];
#pragma unroll
    for (int g = 1; g < 8; ++g) mx = fmaxf(mx, v[g]);
    mx = waveMax32(mx);
    float ex[8], sum = 0.0f;
#pragma unroll
    for (int g = 0; g < 8; ++g) { ex[g] = __expf(v[g] - mx); sum += ex[g]; }
    sum = waveSum32(sum);
    float inv = __frcp_rn(sum);
#pragma unroll
    for (int g = 0; g < 8; ++g)
      atomicAdd(&s_prb[g * 32 + lane], ex[g] * inv);

    // --- per-group top-4 (argmax + mask, 4 rounds x 8 groups) ---
    float cand_v = -__builtin_inff();
    int   cand_i = 0;                     // lane l holds pooled slot l = g*4+j
#pragma unroll
    for (int g = 0; g < 8; ++g) {
      float cur = v[g];
#pragma unroll
      for (int j = 0; j < 4; ++j) {
        float m = waveMax32(cur);
        unsigned long long msk = __ballot(cur == m);
        int src = __ffsll(msk) - 1;       // lowest lane wins ties (top_k order)
        if (lane == src) cur = -__builtin_inff();
        if (lane == (g * 4 + j)) { cand_v = m; cand_i = g * 32 + src; }
      }
    }

    // --- global top-8 over the 32 pooled candidates ---
    float cur = cand_v;
    float wsum = 0.0f, myv = 0.0f;
    int   myi = 0;
#pragma unroll
    for (int j = 0; j < 8; ++j) {
      float m = waveMax32(cur);
      unsigned long long msk = __ballot(cur == m);
      int src = __ffsll(msk) - 1;
      int ei = __shfl(cand_i, src, 32);
      if (lane == src) cur = -__builtin_inff();
      wsum += m;
      if (lane == j) { myv = m; myi = ei; }
    }
    float invw = 1.0f / (wsum + 1e-20f);
    if (lane < 8) {
      out_idx[(size_t)tok * 8 + lane] = (float)myi;   // indices as float output
      out_w[(size_t)tok * 8 + lane]   = myv * invw;
      atomicAdd(&s_cnt[myi], 1.0f);
    }
  }
  __syncthreads();
  atomicAdd(&g_cnt[tid], s_cnt[tid]);
  atomicAdd(&g_prb[tid], s_prb[tid]);
}

// ---------------------------------------------------------------------------
// Kernel 3: aux loss = alpha * sum_e (cnt[e]/(N*K)) * (prb[e]/N)
// ---------------------------------------------------------------------------
__global__ __launch_bounds__(256) void moe_aux_reduce(
    const float* __restrict__ g_cnt, const float* __restrict__ g_prb,
    float* __restrict__ aux, int n_tokens) {
  __shared__ float red[NEXP];
  int e = threadIdx.x;
  float c = g_cnt[e] / ((float)n_tokens * 8.0f);
  float p = g_prb[e] / (float)n_tokens;
  red[e] = c * p;
  __syncthreads();
#pragma unroll
  for (int s = 128; s > 0; s >>= 1) {
    if (e < s) red[e] += red[e + s];
    __syncthreads();
  }
  if (e == 0) aux[0] = red[0] * 0.001f;
}

__global__ void moe_zero_accum(float* __restrict__ p) {
  p[threadIdx.x] = 0.0f;   // 512 floats: cnt[256] ++ prb[256]
}

// ---------------------------------------------------------------------------
extern "C" void kernel_launch(void* const* d_in, const int* in_sizes, int n_in,
                              void* d_out, int out_size, void* d_ws,
                              size_t ws_size, hipStream_t stream) {
  const float* hidden = (const float*)d_in[0];
  const float* weight = (const float*)d_in[1];
  const int n_tokens = in_sizes[0] / HDIM;   // 16384

  float* logits = (float*)d_ws;                          // [N,256] fp32
  float* g_cnt  = logits + (size_t)n_tokens * NEXP;      // [256]
  float* g_prb  = g_cnt + NEXP;                          // [256]

  float* out_idx = (float*)d_out;                        // [N,8] (as float)
  float* out_w   = out_idx + (size_t)n_tokens * 8;       // [N,8]
  float* aux     = out_w + (size_t)n_tokens * 8;         // [1]

  moe_zero_accum<<<1, 512, 0, stream>>>(g_cnt);
  moe_gemm_logits<<<n_tokens / BM, 512, 0, stream>>>(hidden, weight, logits);
  moe_gate_topk<<<(n_tokens + 31) / 32, 256, 0, stream>>>(
      logits, out_idx, out_w, g_cnt, g_prb, n_tokens);
  moe_aux_reduce<<<1, 256, 0, stream>>>(g_cnt, g_prb, aux, n_tokens);
}